// ScaledDotProductAttention_64819646431785
// MI455X (gfx1250) — compile-verified
//
#include <hip/hip_runtime.h>
#include <math.h>

typedef __attribute__((ext_vector_type(2))) float v2f;
typedef __attribute__((ext_vector_type(8))) float v8f;

#define B_SZ 32
#define S_SZ 2048
#define D_SZ 128
#define KT   32      // keys per LDS tile
#define QT   16      // q rows per wave
#define NW   4       // waves per block
#define KP   132     // K tile pitch (floats): lane stride -> 32 distinct banks
#define VP   136     // V tile pitch (floats): half-wave rows offset by 16 banks
#define PP   36      // P staging pitch (floats)

#if __has_builtin(__builtin_amdgcn_global_load_async_to_lds_b128)
#define HAVE_ASYNC 1
#else
#define HAVE_ASYNC 0
#endif

#if HAVE_ASYNC
typedef __attribute__((__vector_size__(4 * sizeof(int)))) int vi4;
typedef __attribute__((address_space(1))) vi4 gvi4;
typedef __attribute__((address_space(3))) vi4 lvi4;
__device__ __forceinline__ void async_ld128(const float* g, float* l) {
    __builtin_amdgcn_global_load_async_to_lds_b128((gvi4*)g, (lvi4*)l, 0, 0);
}
__device__ __forceinline__ void wait_async() {
#if __has_builtin(__builtin_amdgcn_s_wait_asynccnt)
    __builtin_amdgcn_s_wait_asynccnt(0);
#else
    asm volatile("s_wait_asynccnt 0x0" ::: "memory");
#endif
}
#endif

__global__ __launch_bounds__(128)
void fa_fwd_f32_wmma(const float* __restrict__ Q, const float* __restrict__ K,
                     const float* __restrict__ V, const int* __restrict__ M,
                     float* __restrict__ O)
{
    __shared__ float sK[2][KT * KP];
    __shared__ float sV[2][KT * VP];
    __shared__ float sP[NW][QT * PP];

    const int tid  = threadIdx.x;
    const int wave = tid >> 5;
    const int lane = tid & 31;
    const int l16  = lane & 15;
    const int hi   = lane >> 4;     // 0 = lanes 0-15, 1 = lanes 16-31
    const int c    = hi * 2;        // K-dim sub-offset in A/B 16x4 layout

    const int b  = blockIdx.y;
    const int q0 = blockIdx.x * (NW * QT) + wave * QT;

    const float* Qb = Q + (size_t)b * S_SZ * D_SZ;
    const float* Kb = K + (size_t)b * S_SZ * D_SZ;
    const float* Vb = V + (size_t)b * S_SZ * D_SZ;
    const int*   Mb = M + (size_t)b * S_SZ * S_SZ;
    float*       Ob = O + (size_t)b * S_SZ * D_SZ;

    // ---- cooperative K/V tile loader (async double-buffered if available) ----
    auto load_tile = [&](int k0, int buf) {
        for (int idx = tid; idx < KT * (D_SZ / 4); idx += 128) {
            const int row = idx >> 5;              // 32 float4 per row
            const int c4  = (idx & 31) * 4;
            const size_t g = (size_t)(k0 + row) * D_SZ + c4;
#if HAVE_ASYNC
            async_ld128(Kb + g, &sK[buf][row * KP + c4]);
            async_ld128(Vb + g, &sV[buf][row * VP + c4]);
#else
            *(float4*)(&sK[buf][row * KP + c4]) = *(const float4*)(Kb + g);
            *(float4*)(&sV[buf][row * VP + c4]) = *(const float4*)(Vb + g);
            if (k0 + KT < S_SZ) {
                __builtin_prefetch(Kb + g + (size_t)KT * D_SZ, 0, 0);
                __builtin_prefetch(Vb + g + (size_t)KT * D_SZ, 0, 0);
            }
#endif
        }
    };

    // ---- Q tile (16 x 128) preloaded into A-matrix register layout ----
    v2f qa[32];
    {
        const float* qrow = Qb + (size_t)(q0 + l16) * D_SZ;
        #pragma unroll
        for (int i = 0; i < 32; ++i) {
            qa[i].x = qrow[4 * i + c];
            qa[i].y = qrow[4 * i + c + 1];
        }
    }

    v8f acc[8];
    #pragma unroll
    for (int d = 0; d < 8; ++d) acc[d] = (v8f){0.f,0.f,0.f,0.f,0.f,0.f,0.f,0.f};

    float mrow[8], lrow[8];
    #pragma unroll
    for (int r = 0; r < 8; ++r) { mrow[r] = -INFINITY; lrow[r] = 0.f; }

    const float scale = 0.08838834764831845f;   // 1/sqrt(128)

    load_tile(0, 0);                 // prologue: tile 0 in flight

    int buf = 0;
    for (int k0 = 0; k0 < S_SZ; k0 += KT, buf ^= 1) {
#if HAVE_ASYNC
        wait_async();                // my async loads for tile k0 done
#endif
        __syncthreads();             // all waves' loads done / prev reads done
        if (k0 + KT < S_SZ) load_tile(k0 + KT, buf ^ 1);   // overlap next copy

        const float* Kt = sK[buf];
        const float* Vt = sV[buf];

        // ---- scores: two 16x16 subtiles, K-dim = D in 4-wide steps,
        //      B-operands preloaded in chunks of 4 to overlap DS with WMMA ----
        v8f s0 = (v8f){0.f,0.f,0.f,0.f,0.f,0.f,0.f,0.f};
        v8f s1 = (v8f){0.f,0.f,0.f,0.f,0.f,0.f,0.f,0.f};
        #pragma unroll
        for (int ii = 0; ii < 32; ii += 4) {
            v2f b0v[4], b1v[4];
            #pragma unroll
            for (int j = 0; j < 4; ++j) {
                b0v[j] = *(const v2f*)(&Kt[(0  + l16) * KP + 4 * (ii + j) + c]);
                b1v[j] = *(const v2f*)(&Kt[(16 + l16) * KP + 4 * (ii + j) + c]);
            }
            #pragma unroll
            for (int j = 0; j < 4; ++j) {
                s0 = __builtin_amdgcn_wmma_f32_16x16x4_f32(false, qa[ii + j], false,
                        b0v[j], (short)0, s0, false, false);
                s1 = __builtin_amdgcn_wmma_f32_16x16x4_f32(false, qa[ii + j], false,
                        b1v[j], (short)0, s1, false, false);
            }
        }

        // ---- scale + mask + online softmax (C-layout: row = r + hi*8) ----
        float rmax[8];
        #pragma unroll
        for (int r = 0; r < 8; ++r) {
            const int qrow = q0 + r + hi * 8;
            const int* mp = Mb + (size_t)qrow * S_SZ + k0 + l16;
            float v0 = s0[r] * scale;
            float v1 = s1[r] * scale;
            if (mp[0]  == 0) v0 = -1e6f;
            if (mp[16] == 0) v1 = -1e6f;
            s0[r] = v0; s1[r] = v1;
            float mx = fmaxf(v0, v1);
            #pragma unroll
            for (int m = 1; m < 16; m <<= 1)
                mx = fmaxf(mx, __shfl_xor(mx, m, 32));
            rmax[r] = mx;
        }
        #pragma unroll
        for (int r = 0; r < 8; ++r) {
            const float mnew  = fmaxf(mrow[r], rmax[r]);
            const float alpha = __expf(mrow[r] - mnew);
            const float e0 = __expf(s0[r] - mnew);
            const float e1 = __expf(s1[r] - mnew);
            float rs = e0 + e1;
            #pragma unroll
            for (int m = 1; m < 16; m <<= 1)
                rs += __shfl_xor(rs, m, 32);
            lrow[r] = lrow[r] * alpha + rs;
            mrow[r] = mnew;
            s0[r] = e0; s1[r] = e1;
            #pragma unroll
            for (int d = 0; d < 8; ++d) acc[d][r] *= alpha;
        }

        // ---- stage P (C layout -> LDS), wave-local, DS in-order per wave ----
        float* Pw = sP[wave];
        #pragma unroll
        for (int r = 0; r < 8; ++r) {
            const int prow = r + hi * 8;
            Pw[prow * PP + 0  + l16] = s0[r];
            Pw[prow * PP + 16 + l16] = s1[r];
        }

        // ---- PV: A = P (16x4 steps from LDS), B = V columns, preloaded ----
        #pragma unroll
        for (int kk = 0; kk < 8; ++kk) {          // 8 x 4 = 32 keys
            const int kbase = kk * 4 + c;
            const v2f a2 = *(const v2f*)(&Pw[l16 * PP + kbase]);
            v2f bv[8];
            #pragma unroll
            for (int d = 0; d < 8; ++d) {
                bv[d].x = Vt[(kbase    ) * VP + d * 16 + l16];
                bv[d].y = Vt[(kbase + 1) * VP + d * 16 + l16];
            }
            #pragma unroll
            for (int d = 0; d < 8; ++d)
                acc[d] = __builtin_amdgcn_wmma_f32_16x16x4_f32(false, a2, false,
                            bv[d], (short)0, acc[d], false, false);
        }
    }

    // ---- epilogue: normalize rows, store (coalesced 64B segments) ----
    #pragma unroll
    for (int r = 0; r < 8; ++r) {
        const float inv = 1.f / lrow[r];
        float* orow = Ob + (size_t)(q0 + r + hi * 8) * D_SZ;
        #pragma unroll
        for (int d = 0; d < 8; ++d)
            orow[d * 16 + l16] = acc[d][r] * inv;
    }
}

extern "C" void kernel_launch(void* const* d_in, const int* in_sizes, int n_in,
                              void* d_out, int out_size, void* d_ws, size_t ws_size,
                              hipStream_t stream) {
    const float* Q = (const float*)d_in[0];
    const float* K = (const float*)d_in[1];
    const float* V = (const float*)d_in[2];
    const int*   M = (const int*)d_in[3];
    float*       O = (float*)d_out;

    dim3 grid(S_SZ / (NW * QT), B_SZ);   // (32, 32)
    dim3 block(128);
    fa_fwd_f32_wmma<<<grid, block, 0, stream>>>(Q, K, V, M, O);
}